// KMeansClustering_19078244729271
// MI455X (gfx1250) — compile-verified
//
#include <hip/hip_runtime.h>

typedef __attribute__((ext_vector_type(2))) float v2f;
typedef __attribute__((ext_vector_type(8))) float v8f;

#define K_CLUSTERS 16
#define N_FEATURES 256
#define N_POINTS   524288
#define N_ITER     10

#define ASSIGN_BLOCKS   512
#define THREADS         256
#define WAVES_PER_BLOCK 8
#define TILES_TOTAL     (N_POINTS / 16)                                   // 32768
#define TILES_PER_WAVE  (TILES_TOTAL / (ASSIGN_BLOCKS * WAVES_PER_BLOCK)) // 8

// ---------------------------------------------------------------------------
// Init: copy centroids into workspace, compute ||c||^2, zero accumulators.
// ---------------------------------------------------------------------------
__global__ void __launch_bounds__(THREADS)
km_init(const float* __restrict__ c_in, float* __restrict__ c_cur,
        float* __restrict__ cnorm, float* __restrict__ sums,
        float* __restrict__ counts) {
  int t = threadIdx.x;
  for (int i = t; i < K_CLUSTERS * N_FEATURES; i += THREADS) {
    c_cur[i] = c_in[i];
    sums[i]  = 0.0f;
  }
  if (t < K_CLUSTERS) {
    counts[t] = 0.0f;
    float s = 0.0f;
    for (int f = 0; f < N_FEATURES; ++f) {
      float v = c_in[t * N_FEATURES + f];
      s += v * v;
    }
    cnorm[t] = s;
  }
}

// ---------------------------------------------------------------------------
// Fused assign + partial segment-sum kernel.
// Each wave handles 16-point tiles; scores via V_WMMA_F32_16X16X4_F32.
// ---------------------------------------------------------------------------
__global__ void __launch_bounds__(THREADS)
km_assign(const float* __restrict__ x, const float* __restrict__ c_cur,
          const float* __restrict__ cnorm, float* __restrict__ sums_g,
          float* __restrict__ counts_g, float* __restrict__ assign_out) {
  __shared__ float lds_c[K_CLUSTERS * N_FEATURES];     // 16 KB centroid stage
  __shared__ float lds_sums[K_CLUSTERS * N_FEATURES];  // 16 KB block partials
  __shared__ float lds_counts[K_CLUSTERS];

  const int t = threadIdx.x;

  // Stage centroids into LDS, zero the block-local accumulators.
  for (int i = t * 4; i < K_CLUSTERS * N_FEATURES; i += THREADS * 4) {
    *(float4*)&lds_c[i]    = *(const float4*)&c_cur[i];
    *(float4*)&lds_sums[i] = make_float4(0.f, 0.f, 0.f, 0.f);
  }
  if (t < K_CLUSTERS) lds_counts[t] = 0.0f;
  __syncthreads();

  const int lane = t & 31;
  const int wave = t >> 5;
  const int half = lane >> 4;   // 0: K={0,1}/rows 0-7,  1: K={2,3}/rows 8-15
  const int l16  = lane & 15;   // A: row index, B: cluster column, C: column n
  const int gw   = blockIdx.x * WAVES_PER_BLOCK + wave;

  const float cn = cnorm[l16];  // ||c_n||^2 for this lane's cluster column

  for (int tt = 0; tt < TILES_PER_WAVE; ++tt) {
    const int tile = gw * TILES_PER_WAVE + tt;
    const int row0 = tile * 16;

    // A operand: x[row0+l16][kc*4 + 2*half + {0,1}]
    const float* xrow = x + (size_t)(row0 + l16) * N_FEATURES + 2 * half;
    // B operand: c[l16][kc*4 + 2*half + {0,1}] from LDS
    const float* crow = lds_c + l16 * N_FEATURES + 2 * half;

    v8f acc = {};
#pragma unroll
    for (int kc = 0; kc < N_FEATURES / 4; ++kc) {
      v2f a = *(const v2f*)(xrow + kc * 4);
      v2f b = *(const v2f*)(crow + kc * 4);
      acc = __builtin_amdgcn_wmma_f32_16x16x4_f32(
          /*neg_a=*/false, a, /*neg_b=*/false, b,
          /*c_mod=*/(short)0, acc, /*reuse_a=*/false, /*reuse_b=*/false);
    }

    // score(m, n) = ||c_n||^2 - 2 * dot(x_m, c_n).
    // Lane holds n = l16; VGPR r holds point m = half*8 + r.
    float s[8];
    int   bn[8];
#pragma unroll
    for (int r = 0; r < 8; ++r) {
      s[r]  = cn - 2.0f * acc[r];
      bn[r] = l16;
    }

    // Butterfly argmin across the 16 lanes of each half (first-min ties).
#pragma unroll
    for (int m = 1; m < 16; m <<= 1) {
#pragma unroll
      for (int r = 0; r < 8; ++r) {
        float os = __shfl_xor(s[r], m, 32);
        int   on = __shfl_xor(bn[r], m, 32);
        if (os < s[r] || (os == s[r] && on < bn[r])) {
          s[r]  = os;
          bn[r] = on;
        }
      }
    }

    // Write assignments (harness output dtype: float).
    if (l16 == 0) {
#pragma unroll
      for (int r = 0; r < 8; ++r)
        assign_out[row0 + half * 8 + r] = (float)bn[r];
    }

    // Accumulate x into block-local segment sums. Point p's winner lives on
    // lane 0 (p<8) or lane 16 (p>=8) in accumulator slot p&7.
#pragma unroll
    for (int p = 0; p < 16; ++p) {
      const int an = __shfl(bn[p & 7], (p < 8) ? 0 : 16, 32);
      const float* xr  = x + (size_t)(row0 + p) * N_FEATURES + lane * 8;
      float*       dst = lds_sums + an * N_FEATURES + lane * 8;
      float4 v0 = *(const float4*)(xr);
      float4 v1 = *(const float4*)(xr + 4);
      atomicAdd(&dst[0], v0.x);
      atomicAdd(&dst[1], v0.y);
      atomicAdd(&dst[2], v0.z);
      atomicAdd(&dst[3], v0.w);
      atomicAdd(&dst[4], v1.x);
      atomicAdd(&dst[5], v1.y);
      atomicAdd(&dst[6], v1.z);
      atomicAdd(&dst[7], v1.w);
      if (lane == p) atomicAdd(&lds_counts[an], 1.0f);
    }
  }

  __syncthreads();

  // Flush block partials to global with f32 atomics.
  for (int i = t; i < K_CLUSTERS * N_FEATURES; i += THREADS)
    atomicAdd(&sums_g[i], lds_sums[i]);
  if (t < K_CLUSTERS) atomicAdd(&counts_g[t], lds_counts[t]);
}

// ---------------------------------------------------------------------------
// Centroid update: c = count>0 ? sums/max(count,1) : c; recompute ||c||^2;
// rezero accumulators for the next iteration.
// ---------------------------------------------------------------------------
__global__ void __launch_bounds__(THREADS)
km_update(float* __restrict__ c_cur, float* __restrict__ cnorm,
          float* __restrict__ sums, float* __restrict__ counts) {
  __shared__ float red[THREADS];
  const int k = blockIdx.x;
  const int t = threadIdx.x;

  const float cnt = counts[k];
  float v;
  if (cnt > 0.0f)
    v = sums[k * N_FEATURES + t] / fmaxf(cnt, 1.0f);
  else
    v = c_cur[k * N_FEATURES + t];

  c_cur[k * N_FEATURES + t] = v;
  sums[k * N_FEATURES + t]  = 0.0f;

  red[t] = v * v;
  __syncthreads();
  for (int s2 = THREADS / 2; s2 > 0; s2 >>= 1) {
    if (t < s2) red[t] += red[t + s2];
    __syncthreads();
  }
  if (t == 0) {
    cnorm[k]  = red[0];
    counts[k] = 0.0f;
  }
}

// ---------------------------------------------------------------------------
extern "C" void kernel_launch(void* const* d_in, const int* in_sizes, int n_in,
                              void* d_out, int out_size, void* d_ws, size_t ws_size,
                              hipStream_t stream) {
  const float* x  = (const float*)d_in[0];  // [N_POINTS, N_FEATURES]
  const float* c0 = (const float*)d_in[1];  // [K_CLUSTERS, N_FEATURES]
  float* out = (float*)d_out;               // [N_POINTS] assignments (as float)

  float* ws     = (float*)d_ws;
  float* c_cur  = ws;                                   // 4096 floats
  float* cnorm  = ws + K_CLUSTERS * N_FEATURES;         // 16 floats
  float* sums   = cnorm + K_CLUSTERS;                   // 4096 floats
  float* counts = sums + K_CLUSTERS * N_FEATURES;       // 16 floats

  km_init<<<1, THREADS, 0, stream>>>(c0, c_cur, cnorm, sums, counts);
  for (int it = 0; it < N_ITER; ++it) {
    km_assign<<<ASSIGN_BLOCKS, THREADS, 0, stream>>>(x, c_cur, cnorm, sums,
                                                     counts, out);
    km_update<<<K_CLUSTERS, THREADS, 0, stream>>>(c_cur, cnorm, sums, counts);
  }
}